// VGAE_8160437862604
// MI455X (gfx1250) — compile-verified
//
#include <hip/hip_runtime.h>
#include <hip/hip_bf16.h>

typedef float v2f __attribute__((ext_vector_type(2)));
typedef float v8f __attribute__((ext_vector_type(8)));

#define N_NODES 8192
#define N_EDGES 262144
#define IN_F 256
#define HID_F 128
#define OUT_F 64

// ---------------------------------------------------------------------------
// CSR row_ptr via binary search over sorted edge_src (also gives degrees)
// ---------------------------------------------------------------------------
__global__ void rowptr_kernel(const int* __restrict__ src, int* __restrict__ rp) {
    int i = blockIdx.x * blockDim.x + threadIdx.x;
    if (i > N_NODES) return;
    int lo = 0, hi = N_EDGES;
    while (lo < hi) {
        int mid = (lo + hi) >> 1;
        if (src[mid] < i) lo = mid + 1; else hi = mid;
    }
    rp[i] = lo;
}

__global__ void norm_kernel(const int* __restrict__ rp, float* __restrict__ nrm) {
    int i = blockIdx.x * blockDim.x + threadIdx.x;
    if (i >= N_NODES) return;
    int d = rp[i + 1] - rp[i];
    float fd = (float)(d < 1 ? 1 : d);
    nrm[i] = rsqrtf(fd);
}

// ---------------------------------------------------------------------------
// GEMM1: hl1 = (h * norm[:,None]) @ W0^T     [8192x256] x [256x128]
// One 16x16 output tile per wave, f32 WMMA, K stepped by 4.
// ---------------------------------------------------------------------------
__global__ __launch_bounds__(256) void gemm1_kernel(const float* __restrict__ h,
                                                    const float* __restrict__ W0,
                                                    const float* __restrict__ nrm,
                                                    float* __restrict__ hl1) {
    const int tid  = threadIdx.x;
    const int lane = tid & 31;
    const int wave = tid >> 5;
    const int tile = blockIdx.x * 8 + wave;      // 512 blocks * 8 waves = 4096 tiles
    const int tn = tile & 7;                     // 128/16 = 8 N-tiles
    const int tm = tile >> 3;                    // 512 M-tiles
    const int m16  = lane & 15;
    const int half = lane >> 4;

    const int row = tm * 16 + m16;
    const int col = tn * 16 + m16;               // B lane index == N (mod 16)
    const float nm = nrm[row];
    const float* __restrict__ ar = h  + (size_t)row * IN_F;
    const float* __restrict__ br = W0 + (size_t)col * IN_F;  // B[k][n] = W0[n][k]

    v8f acc = {};
    for (int kk = 0; kk < IN_F / 4; ++kk) {
        const int ko = kk * 4 + half * 2;
        v2f a = { ar[ko] * nm, ar[ko + 1] * nm };
        v2f b = { br[ko],      br[ko + 1] };
        acc = __builtin_amdgcn_wmma_f32_16x16x4_f32(false, a, false, b,
                                                    (short)0, acc, false, false);
    }
    for (int r = 0; r < 8; ++r)
        hl1[(size_t)(tm * 16 + r + half * 8) * HID_F + tn * 16 + m16] = acc[r];
}

// ---------------------------------------------------------------------------
// SpMM1 + post-norm + ReLU + fold next layer's pre-norm:
//   h1p[i] = norm[i] * relu(norm[i] * sum_{e: src=i} hl1[dst_e])
// ---------------------------------------------------------------------------
__global__ __launch_bounds__(128) void spmm1_kernel(const float* __restrict__ hl1,
                                                    const int* __restrict__ rp,
                                                    const int* __restrict__ dst,
                                                    const float* __restrict__ nrm,
                                                    float* __restrict__ h1p) {
    const int i = blockIdx.x;
    const int f = threadIdx.x;
    const int s = rp[i], e = rp[i + 1];
    float acc = 0.0f;
    for (int j = s; j < e; ++j)
        acc += hl1[(size_t)dst[j] * HID_F + f];
    const float nm = nrm[i];
    float v = nm * acc;
    v = v > 0.0f ? v : 0.0f;
    h1p[(size_t)i * HID_F + f] = nm * v;
}

// ---------------------------------------------------------------------------
// GEMM2: lincat = h1p @ [Wmu; Wls]^T        [8192x128] x [128x128]
// Columns 0..63 -> Wmu rows, 64..127 -> Wls rows.
// ---------------------------------------------------------------------------
__global__ __launch_bounds__(256) void gemm2_kernel(const float* __restrict__ h1p,
                                                    const float* __restrict__ Wmu,
                                                    const float* __restrict__ Wls,
                                                    float* __restrict__ lincat) {
    const int tid  = threadIdx.x;
    const int lane = tid & 31;
    const int wave = tid >> 5;
    const int tile = blockIdx.x * 8 + wave;      // 4096 tiles
    const int tn = tile & 7;
    const int tm = tile >> 3;
    const int m16  = lane & 15;
    const int half = lane >> 4;

    const int row = tm * 16 + m16;
    const int col = tn * 16 + m16;
    const float* __restrict__ ar = h1p + (size_t)row * HID_F;
    const float* __restrict__ br = (col < OUT_F)
        ? (Wmu + (size_t)col * HID_F)
        : (Wls + (size_t)(col - OUT_F) * HID_F);

    v8f acc = {};
    for (int kk = 0; kk < HID_F / 4; ++kk) {
        const int ko = kk * 4 + half * 2;
        v2f a = { ar[ko], ar[ko + 1] };
        v2f b = { br[ko], br[ko + 1] };
        acc = __builtin_amdgcn_wmma_f32_16x16x4_f32(false, a, false, b,
                                                    (short)0, acc, false, false);
    }
    for (int r = 0; r < 8; ++r)
        lincat[(size_t)(tm * 16 + r + half * 8) * HID_F + tn * 16 + m16] = acc[r];
}

// ---------------------------------------------------------------------------
// SpMM2 + reparameterization:
//   mu = norm[i]*sum lincat[dst][0:64],  ls = norm[i]*sum lincat[dst][64:128]
//   z[i] = mu + exp(ls) * eps[i]
// ---------------------------------------------------------------------------
__global__ __launch_bounds__(64) void spmm2_kernel(const float* __restrict__ lincat,
                                                   const int* __restrict__ rp,
                                                   const int* __restrict__ dst,
                                                   const float* __restrict__ nrm,
                                                   const float* __restrict__ eps,
                                                   float* __restrict__ z) {
    const int i = blockIdx.x;
    const int f = threadIdx.x;   // 0..63
    const int s = rp[i], e = rp[i + 1];
    float am = 0.0f, al = 0.0f;
    for (int j = s; j < e; ++j) {
        const float* __restrict__ p = lincat + (size_t)dst[j] * HID_F;
        am += p[f];
        al += p[OUT_F + f];
    }
    const float nm = nrm[i];
    z[(size_t)i * OUT_F + f] = nm * am + expf(nm * al) * eps[(size_t)i * OUT_F + f];
}

// ---------------------------------------------------------------------------
// GEMM3: logits = z @ z^T   [8192x64] x [64x8192] -> 256 MB fp32 out
// Block = 256 threads (8 waves) computes a 128x128 tile; z panels staged in
// LDS with stride-68 padding (conflict-free, float4-aligned). Each wave owns
// a 16x128 strip = 8 accum tiles. Output streamed with non-temporal stores.
// ---------------------------------------------------------------------------
#define LSTR 68
__global__ __launch_bounds__(256) void zzT_kernel(const float* __restrict__ z,
                                                  float* __restrict__ out) {
    __shared__ float As[128 * LSTR];
    __shared__ float Bs[128 * LSTR];

    const int tid  = threadIdx.x;
    const int lane = tid & 31;
    const int wave = tid >> 5;
    const int m16  = lane & 15;
    const int half = lane >> 4;
    const size_t rowbase = (size_t)blockIdx.y * 128;
    const size_t colbase = (size_t)blockIdx.x * 128;

    // Stage 128x64 row panel and 128x64 col panel (float4, coalesced).
    for (int i = tid; i < 128 * 16; i += 256) {
        const int r  = i >> 4;
        const int c4 = (i & 15) << 2;
        float4 va = *(const float4*)(z + (rowbase + r) * OUT_F + c4);
        float4 vb = *(const float4*)(z + (colbase + r) * OUT_F + c4);
        *(float4*)(As + r * LSTR + c4) = va;
        *(float4*)(Bs + r * LSTR + c4) = vb;
    }
    __syncthreads();

    const float* __restrict__ aRow = As + (wave * 16 + m16) * LSTR;

    v8f acc[8] = {};
    for (int kk = 0; kk < OUT_F / 4; ++kk) {
        const int ko = kk * 4 + half * 2;
        v2f a = *(const v2f*)(aRow + ko);
#pragma unroll
        for (int t = 0; t < 8; ++t) {
            v2f b = *(const v2f*)(Bs + (t * 16 + m16) * LSTR + ko);
            acc[t] = __builtin_amdgcn_wmma_f32_16x16x4_f32(false, a, false, b,
                                                           (short)0, acc[t],
                                                           false, false);
        }
    }

#pragma unroll
    for (int t = 0; t < 8; ++t) {
        const size_t col = colbase + t * 16 + m16;
#pragma unroll
        for (int r = 0; r < 8; ++r) {
            const size_t row = rowbase + wave * 16 + r + half * 8;
            __builtin_nontemporal_store(acc[t][r], out + row * N_NODES + col);
        }
    }
}

// ---------------------------------------------------------------------------
extern "C" void kernel_launch(void* const* d_in, const int* in_sizes, int n_in,
                              void* d_out, int out_size, void* d_ws, size_t ws_size,
                              hipStream_t stream) {
    const float* h        = (const float*)d_in[0];
    const float* W0       = (const float*)d_in[1];
    const float* Wmu      = (const float*)d_in[2];
    const float* Wls      = (const float*)d_in[3];
    const float* eps      = (const float*)d_in[4];
    const int*   edge_src = (const int*)d_in[5];
    const int*   edge_dst = (const int*)d_in[6];
    float* out = (float*)d_out;

    // Workspace carve-up (~10.5 MB)
    char* ws = (char*)d_ws;
    int*   rp   = (int*)ws;    ws += 8448 * sizeof(int);
    float* nrm  = (float*)ws;  ws += (size_t)N_NODES * sizeof(float);
    float* hl1  = (float*)ws;  ws += (size_t)N_NODES * HID_F * sizeof(float); // reused as lincat
    float* h1p  = (float*)ws;  ws += (size_t)N_NODES * HID_F * sizeof(float);
    float* zbuf = (float*)ws;  ws += (size_t)N_NODES * OUT_F * sizeof(float);

    rowptr_kernel<<<(N_NODES + 256) / 256, 256, 0, stream>>>(edge_src, rp);
    norm_kernel<<<N_NODES / 256, 256, 0, stream>>>(rp, nrm);

    gemm1_kernel<<<512, 256, 0, stream>>>(h, W0, nrm, hl1);
    spmm1_kernel<<<N_NODES, 128, 0, stream>>>(hl1, rp, edge_dst, nrm, h1p);

    gemm2_kernel<<<512, 256, 0, stream>>>(h1p, Wmu, Wls, hl1 /* lincat */);
    spmm2_kernel<<<N_NODES, 64, 0, stream>>>(hl1, rp, edge_dst, nrm, eps, zbuf);

    zzT_kernel<<<dim3(64, 64), 256, 0, stream>>>(zbuf, out);
}